// slice_operation_55070070670074
// MI455X (gfx1250) — compile-verified
//
#include <hip/hip_runtime.h>

// Bilateral-grid slice: grid (N=2,C=12,D=8,Hg=16,Wg=16) f32, guide (N,1,1024,1024) f32
// -> out (N,C,1024,1024) f32. Bandwidth-bound: stage grid rows in LDS via CDNA5
// async global->LDS loads, gather from LDS, stream output with NT b128 stores.

#define HH 1024
#define WW 1024
#define CC 12
#define DD 8
#define HG 16
#define WG 16
#define ROWPITCH 17            // 16 + 1 pad (pad[16] = value[15]) -> x-pairs always adjacent

typedef float v4f __attribute__((ext_vector_type(4)));

__global__ __launch_bounds__(256) void bgrid_slice_kernel(
    const float* __restrict__ grid,
    const float* __restrict__ guide,
    float* __restrict__ out)
{
    // [c][d][ysel][x] with x-pitch 17: 12*8*2*17 = 3264 floats = 12.75 KB
    __shared__ float tile[CC * DD * 2 * ROWPITCH];

    const int t = threadIdx.x;
    const int n = blockIdx.x >> 10;          // / HH
    const int y = blockIdx.x & (HH - 1);

    // ---- y interpolation setup (shared by the whole row) ----
    float fy = (float)y * (15.0f / 1023.0f);
    fy = fminf(fy, 15.0f);
    const int   iy0 = (int)fy;
    const float wy  = fy - (float)iy0;
    const int   iy1 = min(iy0 + 1, HG - 1);

    // ---- async-stage the two needed Hg-rows of all (c,d) planes into LDS ----
    // 192 segments of 16 floats (64B) = 768 b128 transfers; 3 per thread.
    const unsigned long long gbase =
        (unsigned long long)(grid + (size_t)n * (CC * DD * HG * WG));
#pragma unroll
    for (int r = 0; r < 3; ++r) {
        const int i    = t + r * 256;        // 0..767
        const int s    = i >> 2;             // segment 0..191 = ((c*8+d)*2+ysel)
        const int q    = i & 3;              // 16-byte quarter within the 64B row
        const int ysel = s & 1;
        const int d    = (s >> 1) & 7;
        const int c    = s >> 4;
        const int yrow = ysel ? iy1 : iy0;
        const unsigned ldsb = (unsigned)((s * ROWPITCH + q * 4) * 4);
        const unsigned goff = (unsigned)((((c * DD + d) * HG + yrow) * WG + q * 4) * 4);
        asm volatile("global_load_async_to_lds_b128 %0, %1, %2"
                     :: "v"(ldsb), "v"(goff), "s"(gbase) : "memory");
    }
    asm volatile("s_wait_asynccnt 0" ::: "memory");
    __syncthreads();
    // fill the pad column so (x0, x0+1) is valid even when ix0 == 15
    if (t < CC * DD * 2) tile[t * ROWPITCH + 16] = tile[t * ROWPITCH + 15];
    __syncthreads();

    // ---- per-pixel setup: 4 consecutive x per thread ----
    const int x4 = t * 4;
    const v4f g4 = *reinterpret_cast<const v4f*>(
        guide + ((size_t)n * HH + y) * WW + x4);

    int   b0[4], b1[4];
    float wxv[4], w00v[4], w01v[4], w10v[4], w11v[4];
    const float gy0 = 1.0f - wy;
#pragma unroll
    for (int j = 0; j < 4; ++j) {
        const int x = x4 + j;
        float fx = (float)x * (15.0f / 1023.0f);
        fx = fminf(fx, 15.0f);
        const int   ix0 = (int)fx;
        wxv[j] = fx - (float)ix0;

        const float g  = g4[j];
        float fz = g * 7.0f;                 // ((g*2-1)+1)*0.5*(D-1)
        fz = fminf(fmaxf(fz, 0.0f), 7.0f);
        const int   iz0 = (int)fz;
        const float wz  = fz - (float)iz0;
        const int   iz1 = min(iz0 + 1, DD - 1);
        const float omz = 1.0f - wz;
        w00v[j] = gy0 * omz;                 // (y0, z0)
        w01v[j] = gy0 * wz;                  // (y0, z1)
        w10v[j] = wy  * omz;                 // (y1, z0)
        w11v[j] = wy  * wz;                  // (y1, z1)
        b0[j] = (iz0 * 2) * ROWPITCH + ix0;  // ysel=0 row of plane d=iz0
        b1[j] = (iz1 * 2) * ROWPITCH + ix0;  // ysel=0 row of plane d=iz1
    }

    // ---- gather + blend + NT store, per channel ----
    float* orow = out + (((size_t)n * CC) * HH + y) * WW + x4;
#pragma unroll
    for (int c = 0; c < CC; ++c) {
        const float* base = &tile[c * (DD * 2 * ROWPITCH)];
        v4f res;
#pragma unroll
        for (int j = 0; j < 4; ++j) {
            const float* p0 = base + b0[j];
            const float* p1 = base + b1[j];
            // x0 corner across 4 (y,z) combos, then x1 corner, then x-lerp
            const float a0 = p0[0] * w00v[j] + p0[ROWPITCH]     * w10v[j]
                           + p1[0] * w01v[j] + p1[ROWPITCH]     * w11v[j];
            const float a1 = p0[1] * w00v[j] + p0[ROWPITCH + 1] * w10v[j]
                           + p1[1] * w01v[j] + p1[ROWPITCH + 1] * w11v[j];
            res[j] = a0 + wxv[j] * (a1 - a0);
        }
        __builtin_nontemporal_store(res, reinterpret_cast<v4f*>(orow));
        orow += (size_t)HH * WW;
    }
}

extern "C" void kernel_launch(void* const* d_in, const int* in_sizes, int n_in,
                              void* d_out, int out_size, void* d_ws, size_t ws_size,
                              hipStream_t stream) {
    (void)in_sizes; (void)n_in; (void)out_size; (void)d_ws; (void)ws_size;
    const float* grid  = (const float*)d_in[0];
    const float* guide = (const float*)d_in[1];
    float* out = (float*)d_out;
    dim3 g(2 * HH);   // one block per (n, row)
    dim3 b(256);      // 8 wave32; each thread covers 4 consecutive x
    hipLaunchKernelGGL(bgrid_slice_kernel, g, b, 0, stream, grid, guide, out);
}